// MultiHeadDQN_86011015069914
// MI455X (gfx1250) — compile-verified
//
#include <hip/hip_runtime.h>
#include <math.h>

#define E_    1024
#define H_    16
#define DH_   64
#define CD_   512
#define S_    2048
#define B_    16
#define SRC_  2049
#define SCPAD 2064
#define BE_   (B_ * E_)
#define SCALE_ 0.125f

typedef float v2f __attribute__((ext_vector_type(2)));
typedef float v8f __attribute__((ext_vector_type(8)));

// ---------------- setup kernels (tiny, fp32 scalar) ----------------

// q_raw[b,e'] = query[b,:] . Wq[e',:] + bq[e']
__global__ void qraw_kernel(const float* __restrict__ q, const float* __restrict__ ipw,
                            const float* __restrict__ ipb, float* __restrict__ qraw) {
  int i = blockIdx.x * 256 + threadIdx.x;           // 16384
  int b = i >> 10, ep = i & 1023;
  const float* w = ipw + (size_t)ep * E_;
  const float* qq = q + (size_t)b * E_;
  float acc = ipb[ep];
  for (int e = 0; e < E_; e += 4) {
    float4 a = *reinterpret_cast<const float4*>(qq + e);
    float4 ww = *reinterpret_cast<const float4*>(w + e);
    acc += a.x * ww.x + a.y * ww.y + a.z * ww.z + a.w * ww.w;
  }
  qraw[i] = acc;
}

// out[b,e'] = [ctx[b], src] . Wc[e',:] + cb[e'], optionally *scale
__global__ void ctxproj_kernel(const float* __restrict__ ctx, const float* __restrict__ src,
                               const float* __restrict__ wc, const float* __restrict__ cb,
                               float* __restrict__ out, float scale, int perB) {
  int i = blockIdx.x * 256 + threadIdx.x;           // 16384
  int b = i >> 10, ep = i & 1023;
  const float* row = wc + (size_t)ep * (CD_ + E_);
  const float* crow = ctx + (size_t)b * CD_;
  float acc = cb[ep];
  for (int j = 0; j < CD_; j += 4) {
    float4 a = *reinterpret_cast<const float4*>(crow + j);
    float4 w = *reinterpret_cast<const float4*>(row + j);
    acc += a.x * w.x + a.y * w.y + a.z * w.z + a.w * w.w;
  }
  const float* srow = perB ? (src + (size_t)b * E_) : src;
  for (int j = 0; j < E_; j += 4) {
    float4 a = *reinterpret_cast<const float4*>(srow + j);
    float4 w = *reinterpret_cast<const float4*>(row + CD_ + j);
    acc += a.x * w.x + a.y * w.y + a.z * w.z + a.w * w.w;
  }
  out[i] = acc * scale;
}

// c0[b,h] = q_fin[b,h,:] . const_b[b,h,:] ; qkl[b,h] = q_fin[b,h,:] . bias_k[h,:]
__global__ void c0_kernel(const float* __restrict__ qfin, const float* __restrict__ constb,
                          const float* __restrict__ bias_k, float* __restrict__ c0,
                          float* __restrict__ qkl) {
  int t = threadIdx.x;                              // 256 = B*H
  int b = t >> 4, h = t & 15;
  float a = 0.f, qk = 0.f;
  for (int d = 0; d < DH_; ++d) {
    float qf = qfin[(size_t)b * E_ + h * DH_ + d];
    a += qf * constb[(size_t)b * E_ + h * DH_ + d];
    qk += qf * bias_k[h * DH_ + d];
  }
  c0[t] = a;
  qkl[t] = qk;
}

// t[b*H+h, j] = sum_d q_fin[b, h*64+d] * Wc_emb[h*64+d, j]
__global__ void t_kernel(const float* __restrict__ qfin, const float* __restrict__ wc,
                         float* __restrict__ t) {
  int r = blockIdx.x;                               // 256 rows
  int b = r >> 4, h = r & 15;
  int j0 = threadIdx.x * 4;
  float4 acc = {0.f, 0.f, 0.f, 0.f};
  for (int d = 0; d < DH_; ++d) {
    float qv = qfin[(size_t)b * E_ + h * DH_ + d];
    float4 w = *reinterpret_cast<const float4*>(wc + (size_t)(h * DH_ + d) * (CD_ + E_) + CD_ + j0);
    acc.x += qv * w.x; acc.y += qv * w.y; acc.z += qv * w.z; acc.w += qv * w.w;
  }
  *reinterpret_cast<float4*>(t + (size_t)r * E_ + j0) = acc;
}

// amat[r, e-quad] = sum_j t[r, j] * Wk[j, e-quad], stored fragment-permuted:
// quad [e0,e1,e2,e3] is written as [e0,e2,e1,e3] so a WMMA B-fragment
// (elements K and K+2) is one aligned b64 load.
__global__ void amat_kernel(const float* __restrict__ t, const float* __restrict__ ipw,
                            float* __restrict__ amat) {
  int r = blockIdx.x;                               // 256 rows
  int e0 = threadIdx.x * 4;
  float4 acc = {0.f, 0.f, 0.f, 0.f};
  for (int j = 0; j < E_; ++j) {
    float tv = t[(size_t)r * E_ + j];
    float4 w = *reinterpret_cast<const float4*>(ipw + (size_t)(E_ + j) * E_ + e0);
    acc.x += tv * w.x; acc.y += tv * w.y; acc.z += tv * w.z; acc.w += tv * w.w;
  }
  float4 perm = {acc.x, acc.z, acc.y, acc.w};       // [e0, e2, e1, e3]
  *reinterpret_cast<float4*>(amat + (size_t)r * E_ + e0) = perm;
}

// ---------------- streaming WMMA kernels (fp32, memory bound) ----------------

// scores[b,h,s] = key[s,b,:] . amat[b,h,:]  via V_WMMA_F32_16X16X4_F32
// A tile = key rows (M=s local, K=e), B tile = permuted amat (K=e, N=h).
// Software-pipelined depth 2 so >=2 loads stay in flight per wave.
__global__ void scores_wmma_kernel(const float* __restrict__ key,
                                   const float* __restrict__ amat,
                                   float* __restrict__ scores) {
  const int tid = threadIdx.x, lane = tid & 31, wave = tid >> 5;
  const int b = blockIdx.y;
  const int s0 = (blockIdx.x * 8 + wave) * 16;
  const int m = lane & 15;
  const int aklo = (lane < 16) ? 0 : 2;             // A: V0->K=aklo, V1->K=aklo+1
  const int boff = (lane < 16) ? 0 : 2;             // permuted B: pair (K, K+2)
  const float* krow = key + (size_t)(s0 + m) * BE_ + (size_t)b * E_ + aklo;
  const float* arow = amat + ((size_t)b * H_ + m) * E_ + boff;
  v8f c = {0.f, 0.f, 0.f, 0.f, 0.f, 0.f, 0.f, 0.f};
  const int NS = E_ / 4;                            // 256 k-steps
  v2f a0 = __builtin_nontemporal_load(reinterpret_cast<const v2f*>(krow));
  v2f b0 = *reinterpret_cast<const v2f*>(arow);
  v2f a1 = __builtin_nontemporal_load(reinterpret_cast<const v2f*>(krow + 4));
  v2f b1 = *reinterpret_cast<const v2f*>(arow + 4);
  for (int k = 0; k + 2 < NS; ++k) {
    v2f a2 = __builtin_nontemporal_load(reinterpret_cast<const v2f*>(krow + 4 * (k + 2)));
    v2f b2 = *reinterpret_cast<const v2f*>(arow + 4 * (k + 2));
    c = __builtin_amdgcn_wmma_f32_16x16x4_f32(false, a0, false, b0, (short)0, c, false, false);
    a0 = a1; b0 = b1; a1 = a2; b1 = b2;
  }
  c = __builtin_amdgcn_wmma_f32_16x16x4_f32(false, a0, false, b0, (short)0, c, false, false);
  c = __builtin_amdgcn_wmma_f32_16x16x4_f32(false, a1, false, b1, (short)0, c, false, false);
  const int n = lane & 15;
  const int mb = (lane < 16) ? 0 : 8;
  float* srow = scores + ((size_t)b * H_ + n) * SCPAD + s0;
#pragma unroll
  for (int r = 0; r < 8; ++r) srow[mb + r] = c[r];
}

// wv[b,h,e] = sum_s w[b,h,s] * value[s,b,e]  via V_WMMA_F32_16X16X4_F32
// A tile = weights (M=h, K=s), B tile = value rows (K=s, N=e local).
__global__ void wv_wmma_kernel(const float* __restrict__ value,
                               const float* __restrict__ w,
                               float* __restrict__ wv) {
  const int tid = threadIdx.x, lane = tid & 31, wave = tid >> 5;
  const int b = blockIdx.y;
  const int e0 = (blockIdx.x * 8 + wave) * 16;
  const int m = lane & 15;
  const int aklo = (lane < 16) ? 0 : 2;
  const int bk = (lane < 16) ? 0 : 1;
  const float* wrow = w + ((size_t)b * H_ + m) * SCPAD + aklo;
  const float* vcol = value + (size_t)(size_t)bk * BE_ + (size_t)b * E_ + e0 + m;
  v8f c = {0.f, 0.f, 0.f, 0.f, 0.f, 0.f, 0.f, 0.f};
  const int NS = S_ / 4;                            // 512 k-steps
  v2f a0 = *reinterpret_cast<const v2f*>(wrow);
  v2f b0; b0[0] = __builtin_nontemporal_load(vcol);
          b0[1] = __builtin_nontemporal_load(vcol + (size_t)2 * BE_);
  v2f a1 = *reinterpret_cast<const v2f*>(wrow + 4);
  v2f b1; b1[0] = __builtin_nontemporal_load(vcol + (size_t)4 * BE_);
          b1[1] = __builtin_nontemporal_load(vcol + (size_t)6 * BE_);
  for (int k = 0; k + 2 < NS; ++k) {
    v2f a2 = *reinterpret_cast<const v2f*>(wrow + 4 * (k + 2));
    v2f b2;
    b2[0] = __builtin_nontemporal_load(vcol + (size_t)(4 * (k + 2)) * BE_);
    b2[1] = __builtin_nontemporal_load(vcol + (size_t)(4 * (k + 2) + 2) * BE_);
    c = __builtin_amdgcn_wmma_f32_16x16x4_f32(false, a0, false, b0, (short)0, c, false, false);
    a0 = a1; b0 = b1; a1 = a2; b1 = b2;
  }
  c = __builtin_amdgcn_wmma_f32_16x16x4_f32(false, a0, false, b0, (short)0, c, false, false);
  c = __builtin_amdgcn_wmma_f32_16x16x4_f32(false, a1, false, b1, (short)0, c, false, false);
  const int n = lane & 15;
  const int mb = (lane < 16) ? 0 : 8;
#pragma unroll
  for (int r = 0; r < 8; ++r)
    wv[((size_t)b * H_ + mb + r) * E_ + e0 + n] = c[r];
}

// ---------------- softmax / average / output ----------------

__global__ void softmax_kernel(float* __restrict__ sc, const float* __restrict__ c0,
                               const float* __restrict__ qkl, float* __restrict__ wl) {
  __shared__ float red[256];
  const int row = blockIdx.x;                       // b*H + h
  const int tid = threadIdx.x;
  float* s = sc + (size_t)row * SCPAD;
  const float c0v = c0[row];
  const float lastv = qkl[row];
  float m = -INFINITY;
  for (int i = tid; i < S_; i += 256) m = fmaxf(m, s[i] + c0v);
  red[tid] = m; __syncthreads();
  for (int st = 128; st > 0; st >>= 1) {
    if (tid < st) red[tid] = fmaxf(red[tid], red[tid + st]);
    __syncthreads();
  }
  const float mx = fmaxf(red[0], lastv);
  __syncthreads();
  float acc = 0.f;
  for (int i = tid; i < S_; i += 256) {
    float v = __expf(s[i] + c0v - mx);
    s[i] = v;
    acc += v;
  }
  red[tid] = acc; __syncthreads();
  for (int st = 128; st > 0; st >>= 1) {
    if (tid < st) red[tid] += red[tid + st];
    __syncthreads();
  }
  const float el = __expf(lastv - mx);
  const float inv = 1.f / (red[0] + el);
  __syncthreads();
  for (int i = tid; i < S_; i += 256) s[i] *= inv;
  if (tid == 0) wl[row] = el * inv;
}

__global__ void avgw_kernel(const float* __restrict__ sc, const float* __restrict__ wl,
                            float* __restrict__ out) {
  int i = blockIdx.x * 256 + threadIdx.x;
  if (i >= B_ * SRC_) return;
  int b = i / SRC_, s = i % SRC_;
  float acc = 0.f;
  if (s < S_) {
    for (int h = 0; h < H_; ++h) acc += sc[(size_t)(b * H_ + h) * SCPAD + s];
  } else {
    for (int h = 0; h < H_; ++h) acc += wl[b * H_ + h];
  }
  out[(size_t)B_ * E_ + i] = acc * (1.f / H_);
}

// attn_pre[b,hd] = wv[b,h,:] . Wv[hd,:] + (1-wl)*bv[hd] + wl*bias_v[hd]
__global__ void attnpre_kernel(const float* __restrict__ wv, const float* __restrict__ ipw,
                               const float* __restrict__ ipb, const float* __restrict__ wl,
                               const float* __restrict__ bias_v, float* __restrict__ pre) {
  int i = blockIdx.x * 256 + threadIdx.x;           // 16384
  int b = i >> 10, hd = i & 1023, h = hd >> 6;
  const float* wrow = ipw + (size_t)(2 * E_ + hd) * E_;
  const float* vv = wv + (size_t)(b * H_ + h) * E_;
  float acc = 0.f;
  for (int e = 0; e < E_; e += 4) {
    float4 a = *reinterpret_cast<const float4*>(vv + e);
    float4 w = *reinterpret_cast<const float4*>(wrow + e);
    acc += a.x * w.x + a.y * w.y + a.z * w.z + a.w * w.w;
  }
  float wlv = wl[b * H_ + h];
  acc += (1.f - wlv) * ipb[2 * E_ + hd] + wlv * bias_v[hd];
  pre[i] = acc;
}

__global__ void outproj_kernel(const float* __restrict__ pre, const float* __restrict__ ow,
                               const float* __restrict__ ob, float* __restrict__ out) {
  int i = blockIdx.x * 256 + threadIdx.x;           // 16384
  int b = i >> 10, o = i & 1023;
  const float* wrow = ow + (size_t)o * E_;
  const float* prow = pre + (size_t)b * E_;
  float acc = ob[o];
  for (int e = 0; e < E_; e += 4) {
    float4 a = *reinterpret_cast<const float4*>(prow + e);
    float4 w = *reinterpret_cast<const float4*>(wrow + e);
    acc += a.x * w.x + a.y * w.y + a.z * w.z + a.w * w.w;
  }
  out[i] = acc;
}

// ---------------- launch ----------------

extern "C" void kernel_launch(void* const* d_in, const int* in_sizes, int n_in,
                              void* d_out, int out_size, void* d_ws, size_t ws_size,
                              hipStream_t stream) {
  const float* query  = (const float*)d_in[0];
  const float* key    = (const float*)d_in[1];
  const float* value  = (const float*)d_in[2];
  const float* ctx    = (const float*)d_in[3];
  const float* ipw    = (const float*)d_in[4];
  const float* ipb    = (const float*)d_in[5];
  const float* ctxw   = (const float*)d_in[6];
  const float* ctxb   = (const float*)d_in[7];
  const float* outw   = (const float*)d_in[8];
  const float* outb   = (const float*)d_in[9];
  const float* bias_k = (const float*)d_in[10];
  const float* bias_v = (const float*)d_in[11];
  float* out = (float*)d_out;

  float* ws = (float*)d_ws;
  // workspace layout (floats)
  float* qfin   = ws;                  // 16384
  float* constb = qfin   + 16384;      // 16384
  float* tbuf   = constb + 16384;      // 262144
  float* amat   = tbuf   + 262144;     // 262144 (fragment-permuted)
  float* c0     = amat   + 262144;     // 256
  float* qkl    = c0     + 256;        // 256
  float* wl     = qkl    + 256;        // 256
  float* scores = wl     + 256;        // 16*16*2064 = 528384
  float* wv     = scores + 528384;     // 262144
  float* pre    = wv     + 262144;     // 16384
  float* qraw   = pre    + 16384;      // 16384

  // 1) q input projection
  qraw_kernel<<<64, 256, 0, stream>>>(query, ipw, ipb, qraw);
  // 2) context injection: q_fin (scaled) and per-batch k constant
  ctxproj_kernel<<<64, 256, 0, stream>>>(ctx, qraw, ctxw, ctxb, qfin, SCALE_, 1);
  ctxproj_kernel<<<64, 256, 0, stream>>>(ctx, ipb + E_, ctxw, ctxb, constb, 1.0f, 0);
  // 3) per-head scalar terms
  c0_kernel<<<1, 256, 0, stream>>>(qfin, constb, bias_k, c0, qkl);
  // 4) A = (q_fin folded through Wc_emb) @ Wk
  t_kernel<<<256, 256, 0, stream>>>(qfin, ctxw, tbuf);
  amat_kernel<<<256, 256, 0, stream>>>(tbuf, ipw, amat);
  // 5) scores: stream key (128 MB) through WMMA
  scores_wmma_kernel<<<dim3(16, 16), 256, 0, stream>>>(key, amat, scores);
  // 6) softmax per (b,h) row (adds c0, handles appended bias_k position)
  softmax_kernel<<<256, 256, 0, stream>>>(scores, c0, qkl, wl);
  // 7) avg_weights output (second tuple element)
  avgw_kernel<<<(B_ * SRC_ + 255) / 256, 256, 0, stream>>>(scores, wl, out);
  // 8) wv: stream value (128 MB) through WMMA
  wv_wmma_kernel<<<dim3(8, 16), 256, 0, stream>>>(value, scores, wv);
  // 9) fold wv through Wv + bias terms, then output projection
  attnpre_kernel<<<64, 256, 0, stream>>>(wv, ipw, ipb, wl, bias_v, pre);
  outproj_kernel<<<64, 256, 0, stream>>>(pre, outw, outb, out);
}